// NLL_15195594293767
// MI455X (gfx1250) — compile-verified
//
#include <hip/hip_runtime.h>
#include <math.h>

// Problem constants (match reference)
#define NOBS 4096
#define Q0G  1000
#define Q1G  500
#define QT   1500      // Q0+Q1
#define PD   1536      // QT padded to multiple of NB (identity padding)
#define NB   64        // Cholesky panel width
#define LOG_2PI 1.837877066409345483560659472811f

typedef __attribute__((ext_vector_type(2))) float v2f;
typedef __attribute__((ext_vector_type(8))) float v8f;

// ---------------------------------------------------------------------------
// 1) Init M = diag(G^-1) padded with identity; zero u, y, x, rr.
// ---------------------------------------------------------------------------
__global__ void nll_init(float* __restrict__ M, float* u, float* y, float* x,
                         float* rr, const float* __restrict__ sig2bs) {
  long long idx = (long long)blockIdx.x * blockDim.x + threadIdx.x;
  const long long total = (long long)PD * PD;
  if (idx < total) {
    int row = (int)(idx / PD), col = (int)(idx % PD);
    float v = 0.0f;
    if (row == col) {
      if (row < Q0G)      v = 1.0f / sig2bs[0];
      else if (row < QT)  v = 1.0f / sig2bs[1];
      else                v = 1.0f;                 // padding -> chol diag 1, log 0
    }
    M[idx] = v;
  }
  if (idx < PD) { u[idx] = 0.0f; y[idx] = 0.0f; x[idx] = 0.0f; }
  if (idx == 0) rr[0] = 0.0f;
}

// ---------------------------------------------------------------------------
// 2) Accumulate M += sig2e^-1 * Z^T Z (histogram), u = Z^T r, rr = r.r
// ---------------------------------------------------------------------------
__global__ void nll_accum(const float* __restrict__ yt, const float* __restrict__ yp,
                          const int* __restrict__ z0, const int* __restrict__ z1,
                          const float* __restrict__ sig2e,
                          float* __restrict__ M, float* __restrict__ u,
                          float* __restrict__ rr) {
  int i = blockIdx.x * blockDim.x + threadIdx.x;
  if (i >= NOBS) return;
  float r = yt[i] - yp[i];
  atomicAdd(rr, r * r);
  float inv_e = 1.0f / sig2e[0];
  int a = z0[i];
  int b = Q0G + z1[i];
  atomicAdd(&M[(long long)a * PD + a], inv_e);
  atomicAdd(&M[(long long)b * PD + b], inv_e);
  atomicAdd(&M[(long long)a * PD + b], inv_e);
  atomicAdd(&M[(long long)b * PD + a], inv_e);
  atomicAdd(&u[a], r);
  atomicAdd(&u[b], r);
}

// ---------------------------------------------------------------------------
// 3a) Factor 64x64 diagonal block in LDS (one block, 64 threads = row each).
// Only lower triangle + diag is read/written meaningfully.
// ---------------------------------------------------------------------------
__global__ void nll_chol_diag(float* __restrict__ M, int kb) {
  __shared__ float s[NB * NB];
  int tid = threadIdx.x;  // 0..63
  for (int k = tid; k < NB * NB; k += 64)
    s[k] = M[(long long)(kb + k / NB) * PD + kb + (k % NB)];
  __syncthreads();
  for (int c = 0; c < NB; ++c) {
    if (tid == 0) s[c * NB + c] = sqrtf(s[c * NB + c]);
    __syncthreads();
    if (tid > c) s[tid * NB + c] /= s[c * NB + c];
    __syncthreads();
    for (int j = c + 1; j <= tid; ++j)               // row tid, cols (c, tid]
      s[tid * NB + j] -= s[tid * NB + c] * s[j * NB + c];
    __syncthreads();
  }
  for (int k = tid; k < NB * NB; k += 64)
    M[(long long)(kb + k / NB) * PD + kb + (k % NB)] = s[k];
}

// ---------------------------------------------------------------------------
// 3b) Panel TRSM: L21 = A21 * L11^-T (forward substitution per row).
// Fully unrolled so the 64-float row stays in VGPRs (no scratch spill).
// ---------------------------------------------------------------------------
__global__ void nll_panel(float* __restrict__ M, int kb, int t0) {
  __shared__ float L[NB * NB];
  int tid = threadIdx.x;
  for (int k = tid; k < NB * NB; k += blockDim.x)
    L[k] = M[(long long)(kb + k / NB) * PD + kb + (k % NB)];
  __syncthreads();
  int r = t0 + blockIdx.x * blockDim.x + tid;
  if (r >= PD) return;
  float a[NB];
  long long base = (long long)r * PD + kb;
#pragma unroll
  for (int c = 0; c < NB; ++c) a[c] = M[base + c];
#pragma unroll
  for (int c = 0; c < NB; ++c) {
    float sum = a[c];
#pragma unroll
    for (int j = 0; j < c; ++j) sum -= a[j] * L[c * NB + j];
    a[c] = sum / L[c * NB + c];
  }
#pragma unroll
  for (int c = 0; c < NB; ++c) M[base + c] = a[c];
}

// ---------------------------------------------------------------------------
// 3c) Trailing SYRK on WMMA: A22 -= L21 * L21^T.
// One wave32 owns a 32x32 region (2x2 register tiles): 4 accumulators,
// 4 x v_wmma_f32_16x16x4_f32 per K-step, 2x operand reuse vs 1 tile/wave.
// A negated in VALU (f32 WMMA supports CNeg only). Lane layouts per ISA 7.12.2.
// ---------------------------------------------------------------------------
__global__ void __launch_bounds__(32)
nll_trailing(float* __restrict__ M, int kb, int t0) {
  int bi = blockIdx.x, bj = blockIdx.y;
  if (bi < bj) return;                     // lower triangle of trailing matrix
  int lane = threadIdx.x;
  int n16  = lane & 15;
  int half = lane >> 4;
  long long i0 = t0 + 32LL * bi;
  long long j0 = t0 + 32LL * bj;

  v8f c00, c01, c10, c11;
#pragma unroll
  for (int v = 0; v < 8; ++v) {
    long long r0 = (i0 + v + 8 * half) * PD;
    long long r1 = (i0 + 16 + v + 8 * half) * PD;
    c00[v] = M[r0 + j0 + n16];
    c01[v] = M[r0 + j0 + 16 + n16];
    c10[v] = M[r1 + j0 + n16];
    c11[v] = M[r1 + j0 + 16 + n16];
  }

  long long arow0 = (i0 + n16) * PD + kb;
  long long arow1 = (i0 + 16 + n16) * PD + kb;
  long long brow0 = (j0 + n16) * PD + kb;
  long long brow1 = (j0 + 16 + n16) * PD + kb;

#pragma unroll
  for (int cc = 0; cc < NB; cc += 4) {
    int ko = cc + 2 * half;
    v2f a0, a1, b0, b1;
    a0.x = -M[arow0 + ko];  a0.y = -M[arow0 + ko + 1];
    a1.x = -M[arow1 + ko];  a1.y = -M[arow1 + ko + 1];
    b0.x =  M[brow0 + ko];  b0.y =  M[brow0 + ko + 1];
    b1.x =  M[brow1 + ko];  b1.y =  M[brow1 + ko + 1];
    // D = (-A)*B + C  => C -= L_i * L_j^T
    c00 = __builtin_amdgcn_wmma_f32_16x16x4_f32(false, a0, false, b0, (short)0, c00, false, false);
    c01 = __builtin_amdgcn_wmma_f32_16x16x4_f32(false, a0, false, b1, (short)0, c01, false, false);
    c10 = __builtin_amdgcn_wmma_f32_16x16x4_f32(false, a1, false, b0, (short)0, c10, false, false);
    c11 = __builtin_amdgcn_wmma_f32_16x16x4_f32(false, a1, false, b1, (short)0, c11, false, false);
  }

#pragma unroll
  for (int v = 0; v < 8; ++v) {
    long long r0 = (i0 + v + 8 * half) * PD;
    long long r1 = (i0 + 16 + v + 8 * half) * PD;
    M[r0 + j0 + n16]      = c00[v];
    M[r0 + j0 + 16 + n16] = c01[v];
    M[r1 + j0 + n16]      = c10[v];
    M[r1 + j0 + 16 + n16] = c11[v];
  }
}

// ---------------------------------------------------------------------------
// 4) Triangular solves L y = u, L^T x = y (single block).
// wave32 shuffle reduction + 8-entry LDS hop: 2 barriers/row (was 9).
// ---------------------------------------------------------------------------
__device__ __forceinline__ float wave_reduce(float v) {
#pragma unroll
  for (int s = 16; s > 0; s >>= 1) v += __shfl_xor(v, s, 32);
  return v;
}

__global__ void nll_fwd_solve(const float* __restrict__ M,
                              const float* __restrict__ u, float* __restrict__ y) {
  __shared__ float red[8];
  int tid  = threadIdx.x;
  int wid  = tid >> 5;
  int lane = tid & 31;
  for (int i = 0; i < PD; ++i) {
    float p = 0.0f;
    for (int j = tid; j < i; j += 256) p += M[(long long)i * PD + j] * y[j];
    p = wave_reduce(p);
    if (lane == 0) red[wid] = p;
    __syncthreads();
    if (tid == 0) {
      float s = red[0] + red[1] + red[2] + red[3]
              + red[4] + red[5] + red[6] + red[7];
      y[i] = (u[i] - s) / M[(long long)i * PD + i];
    }
    __syncthreads();
  }
}

__global__ void nll_bwd_solve(const float* __restrict__ M,
                              const float* __restrict__ y, float* __restrict__ x) {
  __shared__ float red[8];
  int tid  = threadIdx.x;
  int wid  = tid >> 5;
  int lane = tid & 31;
  for (int i = PD - 1; i >= 0; --i) {
    float p = 0.0f;
    for (int j = i + 1 + tid; j < PD; j += 256) p += M[(long long)j * PD + i] * x[j];
    p = wave_reduce(p);
    if (lane == 0) red[wid] = p;
    __syncthreads();
    if (tid == 0) {
      float s = red[0] + red[1] + red[2] + red[3]
              + red[4] + red[5] + red[6] + red[7];
      x[i] = (y[i] - s) / M[(long long)i * PD + i];
    }
    __syncthreads();
  }
}

// ---------------------------------------------------------------------------
// 5) Finalize: logdet V via 2*sum(log L_ii) + lemma terms; quadratic form.
// ---------------------------------------------------------------------------
__global__ void nll_finalize(const float* __restrict__ M, const float* __restrict__ u,
                             const float* __restrict__ x, const float* __restrict__ rr,
                             const float* __restrict__ sig2e,
                             const float* __restrict__ sig2bs,
                             float* __restrict__ out) {
  __shared__ float redA[8], redB[8];
  int tid  = threadIdx.x;
  int wid  = tid >> 5;
  int lane = tid & 31;
  float sl = 0.0f, ux = 0.0f;
  for (int i = tid; i < PD; i += 256) {
    sl += logf(M[(long long)i * PD + i]);     // padding diag == 1 -> log 0
    ux += u[i] * x[i];
  }
  sl = wave_reduce(sl);
  ux = wave_reduce(ux);
  if (lane == 0) { redA[wid] = sl; redB[wid] = ux; }
  __syncthreads();
  if (tid == 0) {
    float sumlog = 0.0f, uTx = 0.0f;
#pragma unroll
    for (int w = 0; w < 8; ++w) { sumlog += redA[w]; uTx += redB[w]; }
    float se = sig2e[0], sb0 = sig2bs[0], sb1 = sig2bs[1];
    float logdetV = 2.0f * sumlog + (float)Q0G * logf(sb0)
                  + (float)Q1G * logf(sb1) + (float)NOBS * logf(se);
    float inv = 1.0f / se;                    // sigma_e^-2 (sig2e is the variance)
    float quad = rr[0] * inv - uTx * inv * inv;   // r^T V^-1 r via Woodbury
    out[0] = 0.5f * (float)NOBS * LOG_2PI + 0.5f * logdetV + 0.5f * quad;
  }
}

// ---------------------------------------------------------------------------
extern "C" void kernel_launch(void* const* d_in, const int* in_sizes, int n_in,
                              void* d_out, int out_size, void* d_ws, size_t ws_size,
                              hipStream_t stream) {
  const float* y_true = (const float*)d_in[0];
  const float* y_pred = (const float*)d_in[1];
  const int*   z0     = (const int*)d_in[2];
  const int*   z1     = (const int*)d_in[3];
  const float* sig2e  = (const float*)d_in[4];
  const float* sig2bs = (const float*)d_in[5];
  float* out = (float*)d_out;

  float* ws = (float*)d_ws;
  float* M  = ws;                                 // PD*PD
  float* u  = M + (size_t)PD * PD;                // PD
  float* yv = u + PD;                             // PD
  float* xv = yv + PD;                            // PD
  float* rr = xv + PD;                            // 1

  // Build M = G^-1 + sig2e^-1 Z^T Z (padded to PD with identity), u = Z^T r.
  const long long mElems = (long long)PD * PD;
  nll_init<<<(unsigned)((mElems + 255) / 256), 256, 0, stream>>>(M, u, yv, xv, rr, sig2bs);
  nll_accum<<<(NOBS + 255) / 256, 256, 0, stream>>>(y_true, y_pred, z0, z1, sig2e, M, u, rr);

  // Blocked right-looking Cholesky of M; trailing SYRK on f32 WMMA (2x2 tiled).
  const int NBLK = PD / NB;
  for (int kbi = 0; kbi < NBLK; ++kbi) {
    int kb = kbi * NB;
    nll_chol_diag<<<1, 64, 0, stream>>>(M, kb);
    int t0 = kb + NB;
    int rem = PD - t0;
    if (rem > 0) {
      nll_panel<<<(rem + 127) / 128, 128, 0, stream>>>(M, kb, t0);
      int tiles32 = rem / 32;
      nll_trailing<<<dim3(tiles32, tiles32), 32, 0, stream>>>(M, kb, t0);
    }
  }

  // Solve M x = u, then assemble scalar NLL.
  nll_fwd_solve<<<1, 256, 0, stream>>>(M, u, yv);
  nll_bwd_solve<<<1, 256, 0, stream>>>(M, yv, xv);
  nll_finalize<<<1, 256, 0, stream>>>(M, u, xv, rr, sig2e, sig2bs, out);
}